// BPRLoss_6597069766796
// MI455X (gfx1250) — compile-verified
//
#include <hip/hip_runtime.h>
#include <hip/hip_bf16.h>

typedef __attribute__((ext_vector_type(2))) float v2f;
typedef __attribute__((ext_vector_type(8))) float v8f;

// Tile kernel: block `blk` owns row-tiles rt1=blk and rt2=T-1-blk (upper
// triangle, perfectly balanced: T+1 column-tiles per block). Each 16x16 tile:
//   diff[m][n] = in_row[m] - in_col[n]      (WMMA, rank-2 A*B)
//   trow[m][n] = t_row[m]                   (WMMA broadcast into C/D layout)
// then a fully branchless softplus + ranking-mask accumulate, counting each
// unordered pair once with the correctly-signed difference:
//   sp(d) and sp(-d) share log1p(exp(-|d|)); only max(+-d,0) differs.
__global__ __launch_bounds__(256) void bpr_tile_kernel(
    const float* __restrict__ input, const float* __restrict__ target,
    float* __restrict__ partials, int T) {
  const int lane = threadIdx.x & 31;
  const int wave = threadIdx.x >> 5;
  const int half = lane >> 4;     // which 16-lane half of the wave
  const int i16  = lane & 15;
  const int blk  = blockIdx.x;

  const int rt1 = blk;
  const int rt2 = T - 1 - blk;
  const int n1  = T - rt1;            // tiles in row-strip rt1
  const int total = n1 + (blk + 1);   // == T + 1 for every block

  const float kLog2e = 1.44269504088896340736f;
  const float kLn2   = 0.69314718055994530942f;

  float acc = 0.0f;

  for (int k = wave; k < total; k += 8) {   // wave-uniform -> EXEC all 1s
    int rt, ct;
    if (k < n1) { rt = rt1; ct = rt1 + k; }
    else        { rt = rt2; ct = rt2 + (k - n1); }
    const int rowBase = rt << 4;
    const int colBase = ct << 4;

    const float in_r = input [rowBase + i16];
    const float t_r  = target[rowBase + i16];
    const float in_c = input [colBase + i16];
    const float t_c  = target[colBase + i16];

    // A (16x4 f32): V0 = K0 (lanes 0-15) / K2 (lanes 16-31); V1 = K1 / K3.
    // Columns: K0 = in_row, K1 = ones, K2 = t_row, K3 = 0.
    v2f a;
    a.x = half ? t_r  : in_r;
    a.y = half ? 0.0f : 1.0f;
    // B_diff (4x16): rows K0 = ones, K1 = -in_col, K2 = K3 = 0.
    v2f b1;
    b1.x = half ? 0.0f : 1.0f;
    b1.y = half ? 0.0f : -in_c;
    // B_trow (4x16): rows K2 = ones, others 0  -> D = broadcast t_row[m].
    v2f b2;
    b2.x = half ? 1.0f : 0.0f;
    b2.y = 0.0f;

    v8f cz = {};
    v8f diff = __builtin_amdgcn_wmma_f32_16x16x4_f32(
        false, a, false, b1, (short)0, cz, false, false);
    v8f trow = __builtin_amdgcn_wmma_f32_16x16x4_f32(
        false, a, false, b2, (short)0, cz, false, false);

    const int jIdx = colBase + i16;
#pragma unroll
    for (int v = 0; v < 8; ++v) {
      const int iIdx = rowBase + v + (half << 3);  // C/D layout: M = v + 8*half
      const float d  = diff[v];
      const float tr = trow[v];
      // shared log-term: log1p(exp(-|d|)) via raw v_exp_f32 / v_log_f32.
      // exp arg <= 0 and log arg in [1,2]: raw instructions are safe.
      const float e  = __builtin_amdgcn_exp2f(-fabsf(d) * kLog2e);
      const float lp = kLn2 * __builtin_amdgcn_logf(1.0f + e);
      // orientation: if t_c > tr the (i,j) pair contributes sp(d), else sp(-d)
      const float sd = (t_c > tr) ? d : -d;
      const float c  = lp + fmaxf(sd, 0.0f);
      // strict upper triangle, ties contribute nothing -- all cndmask, no branch
      const bool take = (t_c != tr) & (jIdx > iIdx);
      acc += take ? c : 0.0f;
    }
  }

  // wave32 reduction
#pragma unroll
  for (int off = 16; off > 0; off >>= 1) acc += __shfl_xor(acc, off, 32);

  __shared__ float wsum[8];
  if (lane == 0) wsum[wave] = acc;
  __syncthreads();
  if (threadIdx.x == 0) {
    float s = 0.0f;
#pragma unroll
    for (int w = 0; w < 8; ++w) s += wsum[w];
    partials[blk] = s;
  }
}

// Deterministic single-block reduction of per-block partials.
__global__ __launch_bounds__(256) void bpr_reduce_kernel(
    const float* __restrict__ partials, int n, float* __restrict__ out,
    float scale) {
  __shared__ float sm[256];
  float s = 0.0f;
  for (int i = threadIdx.x; i < n; i += 256) s += partials[i];
  sm[threadIdx.x] = s;
  __syncthreads();
  for (int off = 128; off > 0; off >>= 1) {
    if ((int)threadIdx.x < off) sm[threadIdx.x] += sm[threadIdx.x + off];
    __syncthreads();
  }
  if (threadIdx.x == 0) out[0] = sm[0] * scale;
}

extern "C" void kernel_launch(void* const* d_in, const int* in_sizes, int n_in,
                              void* d_out, int out_size, void* d_ws,
                              size_t ws_size, hipStream_t stream) {
  (void)n_in; (void)out_size; (void)ws_size;
  const float* input  = (const float*)d_in[0];
  const float* target = (const float*)d_in[1];
  float* out      = (float*)d_out;
  float* partials = (float*)d_ws;

  const int n = in_sizes[0];      // 16384
  const int T = n / 16;           // 1024 tiles per dim
  const int nBlocks = T / 2;      // 512: paired-row triangle balancing

  bpr_tile_kernel<<<nBlocks, 256, 0, stream>>>(input, target, partials, T);

  const float scale = 2.0f / ((float)n * (float)n);
  bpr_reduce_kernel<<<1, 256, 0, stream>>>(partials, nBlocks, out, scale);
}